// e2jmjReverseTransform_38929583571140
// MI455X (gfx1250) — compile-verified
//
#include <hip/hip_runtime.h>

// Reference: 1e6 rows x 64 float32 -> 1 float per row (sign-bit scan).
// HBM-bound: 256 MB read / 23.3 TB/s ~ 11 us floor. No matrix math -> WMMA
// inapplicable; instead use the CDNA5 async global->LDS engine to stream
// 2 KB row-batches per wave (double-buffered, s_wait_asynccnt-ordered),
// wave32 ballots to build the 64-bit sign mask in 2 SGPRs, and scalar clz
// math for the reduction. NT temporal hints keep the one-shot 256 MB stream
// from thrashing the 192 MB L2.

#define BITS 64
#define MVAL 62   // BITS - EXP
#define RPB 8     // rows per async batch (per wave): 8 x 256 B = 2 KB
#define WPB 8     // wave32 per 256-thread block
#define ROWB 256  // bytes per row

static __device__ __forceinline__ unsigned ballot32(bool p) {
  return __builtin_amdgcn_ballot_w32(p);
}

// b0: sign mask of even elements (element 2L in bit L)
// b1: sign mask of odd  elements (element 2L+1 in bit L)
static __device__ __forceinline__ float row_result(unsigned b0, unsigned b1) {
  const bool nz0 = (b0 & 1u) != 0u;  // element 0
  const bool nz1 = (b1 & 1u) != 0u;  // element 1
  // both -> 2, only elem0 -> 3, only elem1 -> 1, none -> 0
  int e = (nz0 | nz1) ? (3 - (nz0 ? 0 : 1) - (nz1 ? 1 : 0)) : 0;

  const unsigned me = b0 & ~1u;  // even elements 2..62
  const unsigned mo = b1 & ~1u;  // odd  elements 3..63
  const int he = me ? 2 * (31 - __builtin_clz(me)) : -1;
  const int ho = mo ? 2 * (31 - __builtin_clz(mo)) + 1 : -1;
  const int lastElem = (he > ho) ? he : ho;  // last nz element, or -1
  const int last_m = lastElem - 2;

  int man;
  if (lastElem < 0) {
    man = 0;
  } else {
    man = (e & 1) ? (MVAL - 1 - last_m) : last_m;
  }
  return (float)(e * MVAL + man);
}

// Issue RPB async row-copies (global -> this wave's LDS staging buffer).
// ISA 08 §4.4: the 24-bit immediate offset is added to BOTH the LDS address
// held in vdst and the global address formed from saddr plus vaddr,
// so per-row we only vary the immediate.
static __device__ __forceinline__ void issue_batch(
    unsigned long long* buf, const unsigned long long* __restrict__ x,
    int batch, int lane) {
  const unsigned ldsBase = (unsigned)(uintptr_t)buf + (unsigned)lane * 8u;
  const unsigned memOff =
      (unsigned)batch * (unsigned)(RPB * ROWB) + (unsigned)lane * 8u;
  asm volatile(
      "global_load_async_to_lds_b64 %0, %1, %2 offset:0 th:TH_LOAD_NT\n\t"
      "global_load_async_to_lds_b64 %0, %1, %2 offset:256 th:TH_LOAD_NT\n\t"
      "global_load_async_to_lds_b64 %0, %1, %2 offset:512 th:TH_LOAD_NT\n\t"
      "global_load_async_to_lds_b64 %0, %1, %2 offset:768 th:TH_LOAD_NT\n\t"
      "global_load_async_to_lds_b64 %0, %1, %2 offset:1024 th:TH_LOAD_NT\n\t"
      "global_load_async_to_lds_b64 %0, %1, %2 offset:1280 th:TH_LOAD_NT\n\t"
      "global_load_async_to_lds_b64 %0, %1, %2 offset:1536 th:TH_LOAD_NT\n\t"
      "global_load_async_to_lds_b64 %0, %1, %2 offset:1792 th:TH_LOAD_NT"
      :
      : "v"(ldsBase), "v"(memOff), "s"(x)
      : "memory");
}

static __device__ __forceinline__ void process_batch(
    const unsigned long long* buf, float* __restrict__ out, int batch,
    int lane) {
  float res[RPB];
#pragma unroll
  for (int r = 0; r < RPB; ++r) {
    const unsigned long long d = buf[r * 32 + lane];  // ds_load_b64
    const float x0 = __uint_as_float((unsigned)(d & 0xFFFFFFFFull));
    const float x1 = __uint_as_float((unsigned)(d >> 32));
    const unsigned b0 = ballot32(x0 >= 0.0f);
    const unsigned b1 = ballot32(x1 >= 0.0f);
    res[r] = row_result(b0, b1);
  }
  if (lane == 0) {
    const int rowBase = batch * RPB;
#pragma unroll
    for (int r = 0; r < RPB; ++r) out[rowBase + r] = res[r];
  }
}

// Direct-load fallback for tail rows (nrows % RPB).
static __device__ __forceinline__ float process_row_direct(
    const unsigned long long* __restrict__ x, int row, int lane) {
  const unsigned long long d =
      __builtin_nontemporal_load(x + (size_t)row * (BITS / 2) + lane);
  const float x0 = __uint_as_float((unsigned)(d & 0xFFFFFFFFull));
  const float x1 = __uint_as_float((unsigned)(d >> 32));
  return row_result(ballot32(x0 >= 0.0f), ballot32(x1 >= 0.0f));
}

__global__ __launch_bounds__(256) void sign_pos_encode_async_kernel(
    const unsigned long long* __restrict__ x, float* __restrict__ out,
    int nrows, int numBatches) {
  __shared__ unsigned long long stage[WPB][2][RPB * 32];  // 32 KB / block

  const int lane = (int)(threadIdx.x & 31);
  const int wib = __builtin_amdgcn_readfirstlane((int)(threadIdx.x >> 5));
  const int wave = __builtin_amdgcn_readfirstlane(
      (int)((blockIdx.x * blockDim.x + threadIdx.x) >> 5));
  const int nwaves = (int)((gridDim.x * blockDim.x) >> 5);

  // Double-buffered async pipeline over full batches (wave-private LDS, so
  // only s_wait_asynccnt ordering is needed -- no barriers).
  if (wave < numBatches) {
    int p = 0;
    issue_batch(stage[wib][0], x, wave, lane);
    for (int b = wave; b < numBatches; b += nwaves) {
      const int nb = b + nwaves;
      if (nb < numBatches) {
        issue_batch(stage[wib][p ^ 1], x, nb, lane);
        // 8 just-issued may be outstanding; previous 8 must be complete.
        asm volatile("s_wait_asynccnt 0x8" ::: "memory");
      } else {
        asm volatile("s_wait_asynccnt 0x0" ::: "memory");
      }
      process_batch(stage[wib][p], out, b, lane);
      p ^= 1;
    }
  }

  // Tail rows via direct NT loads.
  for (int row = numBatches * RPB + wave; row < nrows; row += nwaves) {
    const float r = process_row_direct(x, row, lane);
    if (lane == 0) out[row] = r;
  }
}

extern "C" void kernel_launch(void* const* d_in, const int* in_sizes, int n_in,
                              void* d_out, int out_size, void* d_ws,
                              size_t ws_size, hipStream_t stream) {
  (void)in_sizes;
  (void)n_in;
  (void)d_ws;
  (void)ws_size;
  const unsigned long long* x = (const unsigned long long*)d_in[0];
  float* out = (float*)d_out;

  const int nrows = out_size;  // one output per 64-float row
  const int numBatches = nrows / RPB;

  const int threads = 256;  // 8 wave32 per block
  long long blocksNeeded = ((long long)numBatches + WPB - 1) / WPB;
  int blocks = (int)((blocksNeeded < 2048) ? blocksNeeded : 2048);
  if (blocks < 1) blocks = 1;

  sign_pos_encode_async_kernel<<<blocks, threads, 0, stream>>>(x, out, nrows,
                                                               numBatches);
}